// ScaleDotProductAttention_66322884985196
// MI455X (gfx1250) — compile-verified
//
#include <hip/hip_runtime.h>
#include <stdint.h>

typedef __attribute__((ext_vector_type(2))) float v2f;
typedef __attribute__((ext_vector_type(8))) float v8f;

#define S_LEN 2048
#define D_DIM 64
#define PITCH 2052   // 2048 + 4 floats: breaks LDS bank aliasing between lane halves

// One workgroup = one (b,h) x 16-query-row strip. 8 waves of 32.
extern "C" __global__ __launch_bounds__(256)
void sdpa_softmax_kernel(const float* __restrict__ q,
                         const float* __restrict__ k,
                         const uint8_t* __restrict__ mask,
                         float* __restrict__ out)
{
    extern __shared__ float smem[];   // 16 rows x PITCH floats = 131,328 B

    const int tid  = threadIdx.x;
    const int lane = tid & 31;
    const int wv   = tid >> 5;        // wave id 0..7
    const int hf   = lane >> 4;       // lane half: 0 or 1
    const int l16  = lane & 15;

    const int bh = blockIdx.x >> 7;   // 0..31  (b*16+h)
    const int qt = blockIdx.x & 127;  // query tile 0..127
    const int q0 = qt * 16;

    const float* qbase = q + ((size_t)bh * S_LEN + q0) * D_DIM;
    const float* kbase = k + (size_t)bh * S_LEN * D_DIM;

    // ---------------- Phase 1: S = (Q K^T) * 0.125 via V_WMMA_F32_16X16X4_F32 ----------------
    // f32 A-matrix 16x4 layout: lane L, reg j -> A[M = L&15][K = 2*(L>>4) + j].
    // B-matrix 4x16 is the mirrored layout, which for B = K^T means the SAME
    // per-lane addressing applied to the K matrix. Each fragment = one float2 load.
    v2f a[16];
#pragma unroll
    for (int kk = 0; kk < 16; ++kk)
        a[kk] = *(const v2f*)(qbase + l16 * D_DIM + 4 * kk + 2 * hf);

    for (int t = 0; t < 16; ++t) {
        const int ct = wv + 8 * t;                       // column tile 0..127
        const float* kt = kbase + (size_t)(ct * 16) * D_DIM;
        v2f b[16];
#pragma unroll
        for (int kk = 0; kk < 16; ++kk)
            b[kk] = *(const v2f*)(kt + l16 * D_DIM + 4 * kk + 2 * hf);

        v8f c = {};
#pragma unroll
        for (int kk = 0; kk < 16; ++kk)
            c = __builtin_amdgcn_wmma_f32_16x16x4_f32(false, a[kk], false, b[kk],
                                                      (short)0, c, false, false);

        // C layout: reg r, lane L -> (M = r + 8*(L>>4), N = L&15)
#pragma unroll
        for (int r = 0; r < 8; ++r) {
            const int m = r + 8 * hf;
            smem[m * PITCH + ct * 16 + l16] = c[r] * 0.125f;  // 1/sqrt(64)
        }
    }
    __syncthreads();

    // ---------------- Phase 2: masked row softmax; 2 rows per wave ----------------
    for (int rr = 0; rr < 2; ++rr) {
        const int row = wv * 2 + rr;
        const size_t gro = ((size_t)bh * S_LEN + (size_t)(q0 + row)) * S_LEN;
        const uint8_t* mrow = mask + gro;
        float* orow = out + gro;
        const float* srow = smem + row * PITCH;

        float4 xv[16];
        float mmax = -3.0e38f;
#pragma unroll
        for (int i = 0; i < 16; ++i) {
            const int col = i * 128 + lane * 4;
            float4 x = *(const float4*)(srow + col);
            const uint32_t mk = *(const uint32_t*)(mrow + col);  // 4 bool bytes
            if (mk & 0x000000FFu) x.x = -1.0e9f;
            if (mk & 0x0000FF00u) x.y = -1.0e9f;
            if (mk & 0x00FF0000u) x.z = -1.0e9f;
            if (mk & 0xFF000000u) x.w = -1.0e9f;
            xv[i] = x;
            mmax = fmaxf(mmax, fmaxf(fmaxf(x.x, x.y), fmaxf(x.z, x.w)));
        }
#pragma unroll
        for (int off = 16; off >= 1; off >>= 1)
            mmax = fmaxf(mmax, __shfl_xor(mmax, off, 32));

        float sum = 0.0f;
#pragma unroll
        for (int i = 0; i < 16; ++i) {
            float4 e;
            e.x = __expf(xv[i].x - mmax);
            e.y = __expf(xv[i].y - mmax);
            e.z = __expf(xv[i].z - mmax);
            e.w = __expf(xv[i].w - mmax);
            xv[i] = e;
            sum += (e.x + e.y) + (e.z + e.w);
        }
#pragma unroll
        for (int off = 16; off >= 1; off >>= 1)
            sum += __shfl_xor(sum, off, 32);

        const float inv = 1.0f / sum;
#pragma unroll
        for (int i = 0; i < 16; ++i) {
            const int col = i * 128 + lane * 4;
            float4 o;
            o.x = xv[i].x * inv;
            o.y = xv[i].y * inv;
            o.z = xv[i].z * inv;
            o.w = xv[i].w * inv;
            *(float4*)(orow + col) = o;   // coalesced 512B stores per wave
        }
    }
}

extern "C" void kernel_launch(void* const* d_in, const int* in_sizes, int n_in,
                              void* d_out, int out_size, void* d_ws, size_t ws_size,
                              hipStream_t stream) {
    (void)in_sizes; (void)n_in; (void)out_size; (void)d_ws; (void)ws_size;
    const float*   q    = (const float*)d_in[0];
    const float*   k    = (const float*)d_in[1];
    // d_in[2] = v : unused by the reference (it stops before A@V)
    const uint8_t* mask = (const uint8_t*)d_in[3];
    float*         out  = (float*)d_out;

    const dim3 grid(4096);   // 32 (b*h) * 128 query tiles
    const dim3 block(256);   // 8 wave32
    const size_t ldsBytes = (size_t)16 * PITCH * sizeof(float);  // 131,328 B (<320KB/WGP)
    hipLaunchKernelGGL(sdpa_softmax_kernel, grid, block, ldsBytes, stream,
                       q, k, mask, out);
}